// GNN_43173011259892
// MI455X (gfx1250) — compile-verified
//
#include <hip/hip_runtime.h>

typedef __attribute__((ext_vector_type(2))) float v2f;
typedef __attribute__((ext_vector_type(8))) float v8f;

#define N_NODES 200000
#define N_EDGES 6400000
#define H_DIM   184
#define HT      12                      // 12*16 = 192 >= 184 channel tiles
#define NT      ((N_NODES + 15) / 16)   // 12500 node tiles
#define L1_DIM  188
#define GDIM    200                     // 184 + 16
#define MDH     16

// ---------------------------------------------------------------------------
// 0. zero workspace (agg N*4, deg N, maxbuf 192, hcenter 192) = N*5+384 words
// ---------------------------------------------------------------------------
__global__ __launch_bounds__(256)
void zero_kernel(unsigned* __restrict__ p, int nwords) {
    int i = blockIdx.x * blockDim.x + threadIdx.x;
    if (i < nwords) p[i] = 0u;
}

// ---------------------------------------------------------------------------
// 1. edge scatter: agg[dst] += x[src]*w ; deg[dst] += 1
// ---------------------------------------------------------------------------
__global__ __launch_bounds__(256)
void edge_kernel(const float* __restrict__ nf,
                 const int*   __restrict__ ei,
                 const float* __restrict__ ew,
                 float* __restrict__ agg,
                 float* __restrict__ deg) {
    int e = blockIdx.x * blockDim.x + threadIdx.x;
    if (e >= N_EDGES) return;
    int   src = ei[e];
    int   dst = ei[N_EDGES + e];
    float w   = ew[e];
    const float2* xp = (const float2*)(nf + (size_t)src * 6);  // row stride 6 -> 8B aligned
    float2 x01 = xp[0];
    float2 x23 = xp[1];
    float* ap = agg + (size_t)dst * 4;
    atomicAdd(ap + 0, x01.x * w);
    atomicAdd(ap + 1, x01.y * w);
    atomicAdd(ap + 2, x23.x * w);
    atomicAdd(ap + 3, x23.y * w);
    atomicAdd(deg + dst, 1.0f);
}

// ---------------------------------------------------------------------------
// 2. node phase with V_WMMA_F32_16X16X4_F32:
//    h = relu(mean @ W_rel + b_rel + x @ W_root), fused col-max + center row
//    One wave = 16 nodes; loops over 12 channel tiles; 2 WMMAs per tile.
// ---------------------------------------------------------------------------
__global__ __launch_bounds__(256)
void node_wmma_kernel(const float* __restrict__ nf,
                      const float* __restrict__ agg,
                      const float* __restrict__ deg,
                      const float* __restrict__ Wrel,
                      const float* __restrict__ brel,
                      const float* __restrict__ Wroot,
                      const int*   __restrict__ center,
                      unsigned*    __restrict__ maxbuf,
                      float*       __restrict__ hcenter) {
    __shared__ unsigned maxsh[192];
    int tid = threadIdx.x;
    if (tid < 192) maxsh[tid] = 0u;
    __syncthreads();

    int wave = tid >> 5;
    int lane = tid & 31;
    int tile = blockIdx.x * 8 + wave;
    if (tile < NT) {                         // wave-uniform guard (EXEC stays all-1)
        int n0    = tile * 16;
        int mrow  = lane & 15;
        int node  = n0 + mrow;
        int kbase = (lane >> 4) << 1;        // lanes 0-15: K=0,1 ; lanes 16-31: K=2,3

        float d = fmaxf(deg[node], 1.0f);
        float2 ag = *(const float2*)(agg + (size_t)node * 4 + kbase);
        v2f a_mean; a_mean.x = ag.x / d; a_mean.y = ag.y / d;
        float2 xr = *(const float2*)(nf + (size_t)node * 6 + kbase);
        v2f a_x;    a_x.x = xr.x;       a_x.y = xr.y;

        int ci = *center;
        int mc = ci - n0;                    // center row within this tile (if any)

        for (int ct = 0; ct < HT; ++ct) {
            int  col   = ct * 16 + mrow;
            bool valid = col < H_DIM;
            int  cc    = valid ? col : 0;
            // B 4x16 tile: VGPR v holds K = kbase+v for this lane's half
            v2f b_rel, b_root;
            b_rel.x  = valid ? Wrel [(kbase    ) * H_DIM + cc] : 0.0f;
            b_rel.y  = valid ? Wrel [(kbase + 1) * H_DIM + cc] : 0.0f;
            b_root.x = valid ? Wroot[(kbase    ) * H_DIM + cc] : 0.0f;
            b_root.y = valid ? Wroot[(kbase + 1) * H_DIM + cc] : 0.0f;
            float bias = valid ? brel[cc] : 0.0f;

            v8f c = {};
            c = __builtin_amdgcn_wmma_f32_16x16x4_f32(false, a_mean, false, b_rel,
                                                      (short)0, c, false, false);
            c = __builtin_amdgcn_wmma_f32_16x16x4_f32(false, a_x,    false, b_root,
                                                      (short)0, c, false, false);

            float h[8];
            float m = 0.0f;                  // h >= 0 after relu, 0 is identity
#pragma unroll
            for (int r = 0; r < 8; ++r) {
                h[r] = fmaxf(c[r] + bias, 0.0f);
                m = fmaxf(m, h[r]);
            }
            // rows 0-7 live on lanes 0-15, rows 8-15 on lanes 16-31 (same col)
            m = fmaxf(m, __shfl_xor(m, 16, 32));
            if (lane < 16 && valid)
                atomicMax(&maxsh[col], __float_as_uint(m));   // nonneg: bits monotonic

            if (mc >= 0 && mc < 16) {
                bool mine = (mc < 8) ? (lane < 16) : (lane >= 16);
                float v = 0.0f;
#pragma unroll
                for (int r = 0; r < 8; ++r) if ((mc & 7) == r) v = h[r];
                if (mine && valid) hcenter[col] = v;
            }
        }
    }
    __syncthreads();
    if (tid < 192) {
        unsigned v = maxsh[tid];
        if (v) atomicMax(maxbuf + tid, v);
    }
}

// ---------------------------------------------------------------------------
// 3. head: md MLP (2->16->16), g = [h_c - h_max | md] -> 188 relu -> 5
// ---------------------------------------------------------------------------
__global__ __launch_bounds__(256)
void head_kernel(const float* __restrict__ nf, const int* __restrict__ center,
                 const unsigned* __restrict__ maxbuf, const float* __restrict__ hcenter,
                 const float* __restrict__ W2,  const float* __restrict__ b2,
                 const float* __restrict__ W21, const float* __restrict__ b21,
                 const float* __restrict__ W1,  const float* __restrict__ b1,
                 const float* __restrict__ W4,  const float* __restrict__ b4,
                 float* __restrict__ out) {
    __shared__ float g[GDIM];
    __shared__ float md1[MDH];
    __shared__ float t[L1_DIM];
    int tid = threadIdx.x;
    int ci  = *center;
    if (tid < MDH) {
        float m0 = nf[(size_t)ci * 6 + 4];
        float m1 = nf[(size_t)ci * 6 + 5];
        md1[tid] = fmaxf(m0 * W2[tid] + m1 * W2[MDH + tid] + b2[tid], 0.0f);
    }
    if (tid < H_DIM)
        g[tid] = hcenter[tid] - __uint_as_float(maxbuf[tid]);
    __syncthreads();
    if (tid < MDH) {
        float s = b21[tid];
        for (int k = 0; k < MDH; ++k) s += md1[k] * W21[k * MDH + tid];
        g[H_DIM + tid] = fmaxf(s, 0.0f);
    }
    __syncthreads();
    if (tid < L1_DIM) {
        float s = b1[tid];
        for (int i = 0; i < GDIM; ++i) s += g[i] * W1[i * L1_DIM + tid];
        t[tid] = fmaxf(s, 0.0f);
    }
    __syncthreads();
    if (tid < 5) {
        float s = b4[tid];
        for (int j = 0; j < L1_DIM; ++j) s += t[j] * W4[j * 5 + tid];
        out[tid] = s;
    }
}

// ---------------------------------------------------------------------------
extern "C" void kernel_launch(void* const* d_in, const int* in_sizes, int n_in,
                              void* d_out, int out_size, void* d_ws, size_t ws_size,
                              hipStream_t stream) {
    const float* nf     = (const float*)d_in[0];   // node_feat (N,6)
    const int*   ei     = (const int*)  d_in[1];   // edge_index (2,E)
    const float* ew     = (const float*)d_in[2];   // edge_weight (E)
    const int*   center = (const int*)  d_in[3];   // center_idx scalar
    const float* Wrel   = (const float*)d_in[4];   // (4,184)
    const float* brel   = (const float*)d_in[5];   // (184)
    const float* Wroot  = (const float*)d_in[6];   // (4,184)
    const float* W2     = (const float*)d_in[7];   // (2,16)
    const float* b2     = (const float*)d_in[8];   // (16)
    const float* W21    = (const float*)d_in[9];   // (16,16)
    const float* b21    = (const float*)d_in[10];  // (16)
    const float* W1     = (const float*)d_in[11];  // (200,188)
    const float* b1     = (const float*)d_in[12];  // (188)
    const float* W4     = (const float*)d_in[13];  // (188,5)
    const float* b4     = (const float*)d_in[14];  // (5)
    float* out = (float*)d_out;

    float*    agg     = (float*)d_ws;                  // N*4
    float*    deg     = agg + (size_t)N_NODES * 4;     // N
    unsigned* maxbuf  = (unsigned*)(deg + N_NODES);    // 192
    float*    hcenter = (float*)(maxbuf + 192);        // 192

    int nwords = N_NODES * 5 + 384;
    zero_kernel<<<(nwords + 255) / 256, 256, 0, stream>>>((unsigned*)d_ws, nwords);

    edge_kernel<<<(N_EDGES + 255) / 256, 256, 0, stream>>>(nf, ei, ew, agg, deg);

    node_wmma_kernel<<<(NT + 7) / 8, 256, 0, stream>>>(nf, agg, deg, Wrel, brel, Wroot,
                                                       center, maxbuf, hcenter);

    head_kernel<<<1, 256, 0, stream>>>(nf, center, maxbuf, hcenter,
                                       W2, b2, W21, b21, W1, b1, W4, b4, out);
}